// _DeformableFeatureExtractionLayer_29085518529080
// MI455X (gfx1250) — compile-verified
//
#include <hip/hip_runtime.h>
#include <hip/hip_bf16.h>

typedef float v2f __attribute__((ext_vector_type(2)));
typedef float v8f __attribute__((ext_vector_type(8)));
typedef int   v4i __attribute__((vector_size(4 * sizeof(int))));
typedef __attribute__((address_space(1))) v4i* gvec128_t;
typedef __attribute__((address_space(3))) v4i* lvec128_t;

#define BATCH 16
#define H     104
#define W     68
#define HP    108
#define WP    72
#define KK    5
#define K2    25
#define K2P   28                   // 25 taps padded to multiple of 4
#define S     (H * W)              // 7072
#define NTILE (S / 16)             // 442 N-tiles of 16 pixels per image

// ---------------------------------------------------------------------------
// CDNA5 async global->LDS staging (GLOBAL_LOAD_ASYNC_TO_LDS_B128), with
// portable fallback if the builtin is absent on this toolchain.
// ---------------------------------------------------------------------------
#if defined(__has_builtin)
#if __has_builtin(__builtin_amdgcn_global_load_async_to_lds_b128) && \
    __has_builtin(__builtin_amdgcn_s_wait_asynccnt)
#define USE_ASYNC_LDS 1
#endif
#endif
#ifndef USE_ASYNC_LDS
#define USE_ASYNC_LDS 0
#endif

__device__ __forceinline__ void stage_weights_to_lds(float* lds_dst,
                                                     const float* __restrict__ gsrc,
                                                     int nfloat) {
  int tid = (int)threadIdx.x;
#if USE_ASYNC_LDS
  int nvec = nfloat >> 2;  // 16B chunks
  for (int i = tid; i < nvec; i += 256) {
    __builtin_amdgcn_global_load_async_to_lds_b128(
        (gvec128_t)(gsrc + 4 * i),
        (lvec128_t)(lds_dst + 4 * i),
        0, 0);
  }
  for (int i = (nvec << 2) + tid; i < nfloat; i += 256) lds_dst[i] = gsrc[i];
  __builtin_amdgcn_s_wait_asynccnt(0);
#else
  for (int i = tid; i < nfloat; i += 256) lds_dst[i] = gsrc[i];
#endif
}

// ---------------------------------------------------------------------------
// Replicate-pad (edge mode, 2 each side): [B,C,H,W] -> [B,C,HP,WP]
// ---------------------------------------------------------------------------
__global__ void pad_edge_kernel(const float* __restrict__ in,
                                float* __restrict__ out, int C) {
  int idx = blockIdx.x * blockDim.x + threadIdx.x;
  int total = BATCH * C * HP * WP;
  if (idx >= total) return;
  int wp = idx % WP;
  int t  = idx / WP;
  int hp = t % HP;
  t /= HP;  // b*C + c
  int h = min(max(hp - 2, 0), H - 1);
  int w = min(max(wp - 2, 0), W - 1);
  out[idx] = in[((size_t)t * H + h) * W + w];
}

// ---------------------------------------------------------------------------
// Dense 5x5 VALID conv as implicit GEMM with V_WMMA_F32_16X16X4_F32.
// Weights staged to LDS (async). K iterated as (ci, t0) with tap padded to 28
// so no div/mod appears in the hot loop. One wave = 16 pixels x all channels.
// ---------------------------------------------------------------------------
template <int CI, int MPAD, int MVALID>
__global__ __launch_bounds__(256) void conv_wmma_kernel(
    const float* __restrict__ xp,    // [B][CI][HP][WP]
    const float* __restrict__ wgt,   // [MVALID][CI][5][5]
    const float* __restrict__ bias,  // [MVALID]
    float* __restrict__ out)         // [B][MVALID][H][W]
{
  constexpr int KTOT = CI * K2;
  constexpr int MT   = MPAD / 16;

  __shared__ __align__(16) float w_lds[MVALID * KTOT + 8];
  __shared__ int tapoff[K2P];

  stage_weights_to_lds(w_lds, wgt, MVALID * KTOT);
  if (threadIdx.x < K2P) {
    int t = (int)threadIdx.x;
    tapoff[t] = (t < K2) ? (t / KK) * WP + (t % KK) : 0;
  }
  __syncthreads();

  int wave = (int)((blockIdx.x * blockDim.x + threadIdx.x) >> 5);
  int lane = (int)(threadIdx.x & 31);
  int b   = wave / NTILE;
  int n0  = (wave % NTILE) * 16;
  int ln  = lane & 15;
  int n   = n0 + ln;
  int ho  = n / W, wo = n % W;
  int khi = (lane >= 16) ? 2 : 0;
  int pixbase = ho * WP + wo;

  const float* img = xp + (size_t)b * CI * HP * WP;

  v8f acc[MT];
#pragma unroll
  for (int mt = 0; mt < MT; ++mt)
#pragma unroll
    for (int i = 0; i < 8; ++i) acc[mt][i] = 0.0f;

  for (int ci = 0; ci < CI; ++ci) {
    const float* imgc = img + (size_t)ci * HP * WP;
    if (ci + 1 < CI) __builtin_prefetch(imgc + HP * WP + pixbase, 0, 3);
    int wcol = ci * K2;
#pragma unroll
    for (int t0 = 0; t0 < K2P; t0 += 4) {
      int t = t0 + khi;                  // per-lane tap (t, t+1)
      v2f bfrag;
      {
        float s0 = imgc[pixbase + tapoff[t]];
        float s1 = imgc[pixbase + tapoff[t + 1]];
        bfrag.x = (t < K2) ? s0 : 0.0f;
        bfrag.y = (t + 1 < K2) ? s1 : 0.0f;
      }
#pragma unroll
      for (int mt = 0; mt < MT; ++mt) {
        int m  = mt * 16 + ln;
        int row = (m < MVALID) ? m : 0;
        int tc  = (t < K2 - 1) ? t : (K2 - 2);  // clamp so row stays in range
        float a0 = w_lds[row * KTOT + wcol + tc];
        float a1 = w_lds[row * KTOT + wcol + tc + 1];
        v2f afrag;
        afrag.x = (m < MVALID && t     < K2) ? a0 : 0.0f;
        afrag.y = (m < MVALID && t + 1 < K2) ? a1 : 0.0f;
        acc[mt] = __builtin_amdgcn_wmma_f32_16x16x4_f32(
            false, afrag, false, bfrag, (short)0, acc[mt], false, false);
      }
    }
  }

  float* outb = out + (size_t)b * MVALID * S;
#pragma unroll
  for (int mt = 0; mt < MT; ++mt) {
#pragma unroll
    for (int r = 0; r < 8; ++r) {
      int m = mt * 16 + r + ((lane >= 16) ? 8 : 0);
      if (m < MVALID) outb[(size_t)m * S + n] = acc[mt][r] + bias[m];
    }
  }
}

// ---------------------------------------------------------------------------
// Deformable conv v1 as implicit GEMM with WMMA f32.
// Per-wave LDS table of bilinear samples: for each (tap, pixel) store 4
// clamped corner offsets + 4 validity-masked weights (zeros for taps >= 25),
// so the channel loop has no guards, no div/mod, no offset refetch.
// ---------------------------------------------------------------------------
template <int CI, int MT>
__global__ __launch_bounds__(256) void deform_wmma_kernel(
    const float* __restrict__ xp,    // [B][CI][HP][WP]
    const float* __restrict__ off,   // [B][2*K2][H][W]
    const float* __restrict__ wgt,   // [MT*16][CI][5][5]
    const float* __restrict__ bias,  // [MT*16]
    float* __restrict__ out)         // [B][MT*16][H][W]
{
  constexpr int KTOT = CI * K2;
  constexpr int MVAL = MT * 16;
  constexpr int NENT = K2P * 16;     // 448 entries per wave

  __shared__ __align__(16) unsigned int dtab[8 * NENT * 8];  // 114.7 KB

  int wave = (int)((blockIdx.x * blockDim.x + threadIdx.x) >> 5);
  int wv   = (int)(threadIdx.x >> 5);
  int lane = (int)(threadIdx.x & 31);
  int b   = wave / NTILE;
  int n0  = (wave % NTILE) * 16;
  int ln  = lane & 15;
  int n   = n0 + ln;
  int khi = (lane >= 16) ? 2 : 0;

  const float* img  = xp  + (size_t)b * CI * HP * WP;
  const float* offb = off + (size_t)b * (2 * K2) * S;
  unsigned int* tab = dtab + wv * (NENT * 8);

  // -------- precompute bilinear table (per wave) --------
  for (int e = lane; e < NENT; e += 32) {
    int t   = e >> 4;
    int pix = e & 15;
    int ne  = n0 + pix;
    int hoe = ne / W, woe = ne % W;
    int i00 = 0, i01 = 0, i10 = 0, i11 = 0;
    float w00 = 0.f, w01 = 0.f, w10 = 0.f, w11 = 0.f;
    if (t < K2) {
      float offy = offb[(size_t)(2 * t)     * S + ne];
      float offx = offb[(size_t)(2 * t + 1) * S + ne];
      float y = (float)(hoe + t / KK) + offy;
      float x = (float)(woe + t % KK) + offx;
      float y0f = floorf(y), x0f = floorf(x);
      int   y0 = (int)y0f,   x0 = (int)x0f;
      float wy = y - y0f,    wx = x - x0f;
      int y0c = min(max(y0, 0), HP - 1), y1c = min(max(y0 + 1, 0), HP - 1);
      int x0c = min(max(x0, 0), WP - 1), x1c = min(max(x0 + 1, 0), WP - 1);
      bool vy0 = (y0 >= 0) & (y0 < HP), vy1 = (y0 + 1 >= 0) & (y0 + 1 < HP);
      bool vx0 = (x0 >= 0) & (x0 < WP), vx1 = (x0 + 1 >= 0) & (x0 + 1 < WP);
      i00 = y0c * WP + x0c; i01 = y0c * WP + x1c;
      i10 = y1c * WP + x0c; i11 = y1c * WP + x1c;
      w00 = (vy0 && vx0) ? (1.f - wy) * (1.f - wx) : 0.f;
      w01 = (vy0 && vx1) ? (1.f - wy) * wx         : 0.f;
      w10 = (vy1 && vx0) ? wy * (1.f - wx)         : 0.f;
      w11 = (vy1 && vx1) ? wy * wx                 : 0.f;
    }
    unsigned int* ent = tab + e * 8;
    ent[0] = (unsigned int)i00; ent[1] = (unsigned int)i01;
    ent[2] = (unsigned int)i10; ent[3] = (unsigned int)i11;
    ent[4] = __float_as_uint(w00); ent[5] = __float_as_uint(w01);
    ent[6] = __float_as_uint(w10); ent[7] = __float_as_uint(w11);
  }
  __syncthreads();

  v8f acc[MT];
#pragma unroll
  for (int mt = 0; mt < MT; ++mt)
#pragma unroll
    for (int i = 0; i < 8; ++i) acc[mt][i] = 0.0f;

  for (int ci = 0; ci < CI; ++ci) {
    const float* imgc = img + (size_t)ci * HP * WP;
    if (ci + 1 < CI) __builtin_prefetch(imgc + HP * WP, 0, 3);
    const float* wci = wgt + (size_t)ci * K2;  // + m*KTOT gives row
#pragma unroll
    for (int t0 = 0; t0 < K2P; t0 += 4) {
      int t = t0 + khi;
      v2f bfrag;
      {
        const unsigned int* e0 = tab + (t * 16 + ln) * 8;
        const unsigned int* e1 = tab + ((t + 1) * 16 + ln) * 8;
        bfrag.x = __uint_as_float(e0[4]) * imgc[e0[0]] +
                  __uint_as_float(e0[5]) * imgc[e0[1]] +
                  __uint_as_float(e0[6]) * imgc[e0[2]] +
                  __uint_as_float(e0[7]) * imgc[e0[3]];
        bfrag.y = __uint_as_float(e1[4]) * imgc[e1[0]] +
                  __uint_as_float(e1[5]) * imgc[e1[1]] +
                  __uint_as_float(e1[6]) * imgc[e1[2]] +
                  __uint_as_float(e1[7]) * imgc[e1[3]];
      }
#pragma unroll
      for (int mt = 0; mt < MT; ++mt) {
        int m  = mt * 16 + ln;
        int tc = (t < K2 - 1) ? t : (K2 - 2);   // clamp to stay in bounds
        float a0 = wci[(size_t)m * KTOT + tc];
        float a1 = wci[(size_t)m * KTOT + tc + 1];
        v2f afrag;
        afrag.x = (t     < K2) ? a0 : 0.0f;
        afrag.y = (t + 1 < K2) ? a1 : 0.0f;
        acc[mt] = __builtin_amdgcn_wmma_f32_16x16x4_f32(
            false, afrag, false, bfrag, (short)0, acc[mt], false, false);
      }
    }
  }

  float* outb = out + (size_t)b * MVAL * S;
#pragma unroll
  for (int mt = 0; mt < MT; ++mt) {
#pragma unroll
    for (int r = 0; r < 8; ++r) {
      int m = mt * 16 + r + ((lane >= 16) ? 8 : 0);
      float v = acc[mt][r] + bias[m];
      outb[(size_t)m * S + n] = v > 0.0f ? v : 0.0f;
    }
  }
}

// ---------------------------------------------------------------------------
// Launch: pad -> offset-conv -> deform -> pad -> offset-conv -> deform
// ---------------------------------------------------------------------------
extern "C" void kernel_launch(void* const* d_in, const int* in_sizes, int n_in,
                              void* d_out, int out_size, void* d_ws, size_t ws_size,
                              hipStream_t stream) {
  (void)in_sizes; (void)n_in; (void)out_size; (void)ws_size;

  const float* x     = (const float*)d_in[0];
  const float* oc1_w = (const float*)d_in[1];
  const float* oc1_b = (const float*)d_in[2];
  const float* dc1_w = (const float*)d_in[3];
  const float* dc1_b = (const float*)d_in[4];
  const float* oc2_w = (const float*)d_in[5];
  const float* oc2_b = (const float*)d_in[6];
  const float* dc2_w = (const float*)d_in[7];
  const float* dc2_b = (const float*)d_in[8];
  float* out = (float*)d_out;

  // Workspace layout (floats)
  const size_t XP1_N = (size_t)BATCH * 13 * HP * WP;   // 1,617,408
  const size_t OFF_N = (size_t)BATCH * 50 * S;         // 5,657,600 (off1/off2 shared)
  const size_t Hm_N  = (size_t)BATCH * 32 * S;         // 3,620,864
  float* ws   = (float*)d_ws;
  float* xp1  = ws;
  float* offb = xp1 + XP1_N;
  float* hbuf = offb + OFF_N;
  float* hp   = hbuf + Hm_N;

  const int gemm_blocks = (BATCH * NTILE * 32) / 256;  // 884

  {
    int total = BATCH * 13 * HP * WP;
    pad_edge_kernel<<<(total + 255) / 256, 256, 0, stream>>>(x, xp1, 13);
  }
  conv_wmma_kernel<13, 64, 50><<<gemm_blocks, 256, 0, stream>>>(xp1, oc1_w, oc1_b, offb);
  deform_wmma_kernel<13, 2><<<gemm_blocks, 256, 0, stream>>>(xp1, offb, dc1_w, dc1_b, hbuf);
  {
    int total = BATCH * 32 * HP * WP;
    pad_edge_kernel<<<(total + 255) / 256, 256, 0, stream>>>(hbuf, hp, 32);
  }
  conv_wmma_kernel<32, 64, 50><<<gemm_blocks, 256, 0, stream>>>(hp, oc2_w, oc2_b, offb);
  deform_wmma_kernel<32, 4><<<gemm_blocks, 256, 0, stream>>>(hp, offb, dc2_w, dc2_b, out);
}